// DSNAS_v_55216099558218
// MI455X (gfx1250) — compile-verified
//
#include <hip/hip_runtime.h>
#include <cstdint>
#include <cstddef>

typedef __attribute__((ext_vector_type(2))) float v2f;
typedef __attribute__((ext_vector_type(8))) float v8f;

#define NCOLS  8
#define NEMB   12
#define DD     128
#define BB     8192
#define NPAIR  28
#define NENT   56
#define ECHUNK 8
#define EPC    7   // entries per grid.y chunk (8*7 = 56)

__device__ __forceinline__ int argmax5(const float* __restrict__ la) {
  int pos = 0;
  float best = la[0];
#pragma unroll
  for (int t = 1; t < 5; ++t) {
    float v = la[t];
    if (v > best) { best = v; pos = t; }
  }
  return pos;
}

// ---------------------------------------------------------------------------
// Prep: softplus table, per-pair routing (argmax), task list + weight copies.
// Entry 2k  : mode 0..3 (add/mul/max/min with W_nc[k,pos]) or mode 4 (p @ Wcat[:, :128])
// Entry 2k+1: mode 5 (q @ Wcat[:, 128:]) if pos==4, else invalid (255)
// ---------------------------------------------------------------------------
__global__ __launch_bounds__(256) void dsnas_prep(
    const float* __restrict__ emb_std, const float* __restrict__ W_nc,
    const float* __restrict__ W_cat, const float* __restrict__ log_alpha,
    float* __restrict__ Ssp, int4* __restrict__ meta, float* __restrict__ Wsel) {
  const int t = blockIdx.x * blockDim.x + threadIdx.x;  // 16384 threads

  if (t < NCOLS * NEMB * DD) {
    float x = emb_std[t];
    Ssp[t] = (x > 20.f) ? x : log1pf(__expf(x));
  }

  if (t < NENT * 2 * DD) {  // one thread per selected-weight element
    const int e = t >> 8, dn = t & 255, n = dn >> 7, d = dn & 127;
    const int k = e >> 1, half = e & 1;
    const int pos = argmax5(log_alpha + k * 5);
    if (half == 0) {
      Wsel[t] = (pos < 4) ? W_nc[(((k * 4 + pos) * 2 + n) << 7) + d]
                          : W_cat[(k * 2 + n) * 256 + d];
    } else if (pos == 4) {
      Wsel[t] = W_cat[(k * 2 + n) * 256 + 128 + d];
    }
  }

  if (t < NPAIR) {
    const int k = t;
    const int pos = argmax5(log_alpha + k * 5);
    int pi = 0, pj = 1, cnt = 0;
    for (int a = 0; a < NCOLS; ++a)
      for (int c2 = a + 1; c2 < NCOLS; ++c2) {
        if (cnt == k) { pi = a; pj = c2; }
        ++cnt;
      }
    meta[2 * k]     = make_int4(pi, pj, (pos < 4) ? pos : 4, 0);
    meta[2 * k + 1] = make_int4(pi, pj, (pos == 4) ? 5 : 255, 0);
  }
}

// ---------------------------------------------------------------------------
// One task entry: 32 chained V_WMMA_F32_16X16X4_F32 over K=128.
// Lane (r = lane&15, h = lane>>4): A/B both consume d = 4c+2h, 4c+2h+1.
// ---------------------------------------------------------------------------
template <int MODE>
__device__ __forceinline__ v8f entry_accum(
    v8f C, const float* __restrict__ Mi, const float* __restrict__ Si,
    const float* __restrict__ ni, const float* __restrict__ Mj,
    const float* __restrict__ Sj, const float* __restrict__ nj,
    const float* __restrict__ W, int r, int h) {
#pragma unroll
  for (int c = 0; c < 32; ++c) {
    const int d = (c << 2) + (h << 1);
    v2f p, q, a;
    if (MODE != 5) {
      v2f mi = *(const v2f*)(Mi + d);
      v2f si = *(const v2f*)(Si + d);
      v2f xi = *(const v2f*)(ni + d);
      p.x = fmaf(si.x * 0.01f, xi.x, mi.x);
      p.y = fmaf(si.y * 0.01f, xi.y, mi.y);
    }
    if (MODE <= 3 || MODE == 5) {
      v2f mj = *(const v2f*)(Mj + d);
      v2f sj = *(const v2f*)(Sj + d);
      v2f xj = *(const v2f*)(nj + d);
      q.x = fmaf(sj.x * 0.01f, xj.x, mj.x);
      q.y = fmaf(sj.y * 0.01f, xj.y, mj.y);
    }
    if      (MODE == 0) { a.x = p.x + q.x;      a.y = p.y + q.y; }
    else if (MODE == 1) { a.x = p.x * q.x;      a.y = p.y * q.y; }
    else if (MODE == 2) { a.x = fmaxf(p.x, q.x); a.y = fmaxf(p.y, q.y); }
    else if (MODE == 3) { a.x = fminf(p.x, q.x); a.y = fminf(p.y, q.y); }
    else if (MODE == 4) { a = p; }
    else                { a = q; }

    v2f bm = {0.f, 0.f};
    if (r < 2) bm = *(const v2f*)(W + (r << 7) + d);  // B[K=2h(+1)][N=r]

    C = __builtin_amdgcn_wmma_f32_16x16x4_f32(
        /*neg_a=*/false, a, /*neg_b=*/false, bm,
        /*c_mod=*/(short)0, C, /*reuse_a=*/false, /*reuse_b=*/false);
  }
  return C;
}

__global__ __launch_bounds__(256) void dsnas_main(
    const int* __restrict__ features, const float* __restrict__ emb_mean,
    const float* __restrict__ noise, const float* __restrict__ Ssp,
    const int4* __restrict__ meta, const float* __restrict__ Wsel,
    float* __restrict__ partial) {
  const int lane = threadIdx.x & 31;
  const int wave = threadIdx.x >> 5;
  const int r = lane & 15, h = lane >> 4;
  const int btile = blockIdx.x * 128 + wave * 16;  // 16 batch rows per wave
  const int b = btile + r;
  const int g = blockIdx.y;

  v8f C = {0.f, 0.f, 0.f, 0.f, 0.f, 0.f, 0.f, 0.f};

  const int e0 = g * EPC;
  for (int e = e0; e < e0 + EPC; ++e) {
    const int4 m4 = meta[e];
    const int mode = __builtin_amdgcn_readfirstlane(m4.z);
    if (mode > 5) continue;  // uniform scalar branch, EXEC untouched
    const int ci = __builtin_amdgcn_readfirstlane(m4.x);
    const int cj = __builtin_amdgcn_readfirstlane(m4.y);
    const int k = e >> 1;

    const int fi = features[ci * BB + b];
    const int fj = features[cj * BB + b];
    const float* Mi = emb_mean + ((ci * NEMB + fi) << 7);
    const float* Si = Ssp      + ((ci * NEMB + fi) << 7);
    const float* Mj = emb_mean + ((cj * NEMB + fj) << 7);
    const float* Sj = Ssp      + ((cj * NEMB + fj) << 7);
    const float* ni = noise + (((long)(k * 2 + 0) * BB + b) << 7);
    const float* nj = noise + (((long)(k * 2 + 1) * BB + b) << 7);
    const float* W = Wsel + (e << 8);

    switch (mode) {
      case 0: C = entry_accum<0>(C, Mi, Si, ni, Mj, Sj, nj, W, r, h); break;
      case 1: C = entry_accum<1>(C, Mi, Si, ni, Mj, Sj, nj, W, r, h); break;
      case 2: C = entry_accum<2>(C, Mi, Si, ni, Mj, Sj, nj, W, r, h); break;
      case 3: C = entry_accum<3>(C, Mi, Si, ni, Mj, Sj, nj, W, r, h); break;
      case 4: C = entry_accum<4>(C, Mi, Si, ni, Mj, Sj, nj, W, r, h); break;
      default: C = entry_accum<5>(C, Mi, Si, ni, Mj, Sj, nj, W, r, h); break;
    }
  }

  // C/D layout: lane n (<2) holds rows m = 8h+vv, column n = r.
  if (r < 2) {
#pragma unroll
    for (int vv = 0; vv < 8; ++vv) {
      const int m = h * 8 + vv;
      partial[(size_t)g * (BB * 2) + (size_t)(btile + m) * 2 + r] = C[vv];
    }
  }
}

__global__ __launch_bounds__(256) void dsnas_reduce(
    const float* __restrict__ partial, float* __restrict__ out) {
  const int idx = blockIdx.x * blockDim.x + threadIdx.x;
  if (idx < BB * 2) {
    float s = 0.f;
#pragma unroll
    for (int gg = 0; gg < ECHUNK; ++gg) s += partial[(size_t)gg * (BB * 2) + idx];
    out[idx] = s;
  }
}

extern "C" void kernel_launch(void* const* d_in, const int* in_sizes, int n_in,
                              void* d_out, int out_size, void* d_ws, size_t ws_size,
                              hipStream_t stream) {
  const int*   features  = (const int*)d_in[0];
  const float* emb_mean  = (const float*)d_in[1];
  const float* emb_std   = (const float*)d_in[2];
  const float* W_nc      = (const float*)d_in[3];
  const float* W_cat     = (const float*)d_in[4];
  const float* log_alpha = (const float*)d_in[5];
  const float* noise     = (const float*)d_in[6];
  float* out = (float*)d_out;

  // Workspace carve-up (all 16B-aligned):
  float* Ssp  = (float*)d_ws;                       // 12288 f
  int4*  meta = (int4*)(Ssp + NCOLS * NEMB * DD);   // 56 int4
  float* Wsel = (float*)(meta + NENT);              // 56*256 f
  float* partial = Wsel + NENT * 2 * DD;            // 8*8192*2 f

  dsnas_prep<<<64, 256, 0, stream>>>(emb_std, W_nc, W_cat, log_alpha, Ssp, meta, Wsel);
  dsnas_main<<<dim3(BB / 128, ECHUNK), 256, 0, stream>>>(features, emb_mean, noise,
                                                         Ssp, meta, Wsel, partial);
  dsnas_reduce<<<(BB * 2 + 255) / 256, 256, 0, stream>>>(partial, out);
}